// Projector_38414187496061
// MI455X (gfx1250) — compile-verified
//
#include <hip/hip_runtime.h>

// Problem constants (from reference)
#define NUM_FROM 524288
#define NUM_TO   262144
#define NNZ_EXP  1048576
#define D        128

// CSR-build geometry: NUM_TO = 1024 blocks * 256 elements
#define SCAN_BLOCKS 1024
#define SCAN_CHUNK  256

typedef unsigned int uint32;

// ===========================================================================
// Phase 0: zero the histogram counts (uint4 stores -> global_store_b128)
// ===========================================================================
__global__ __launch_bounds__(256) void zero_u32_kernel(uint4* __restrict__ p,
                                                       int n4) {
    int i = blockIdx.x * blockDim.x + threadIdx.x;
    if (i < n4) p[i] = make_uint4(0u, 0u, 0u, 0u);
}

// ===========================================================================
// Phase 1: histogram of to_idx  (1M cheap u32 atomics, coalesced index reads)
// ===========================================================================
__global__ __launch_bounds__(256) void hist_kernel(const int* __restrict__ to_idx,
                                                   uint32* __restrict__ counts,
                                                   int nnz) {
    int e = blockIdx.x * blockDim.x + threadIdx.x;
    if (e < nnz) {
        __builtin_prefetch(&to_idx[e + blockDim.x * gridDim.x], 0, 3);
        atomicAdd(&counts[to_idx[e]], 1u);
    }
}

// ===========================================================================
// Phase 2a: per-block exclusive scan of 256 counts; emit block totals.
// ===========================================================================
__global__ __launch_bounds__(SCAN_CHUNK) void block_scan_kernel(
    const uint32* __restrict__ counts,
    uint32* __restrict__ offsets,
    uint32* __restrict__ blockSums) {
    __shared__ uint32 s[SCAN_CHUNK];
    const int tid = threadIdx.x;
    const int i = blockIdx.x * SCAN_CHUNK + tid;
    uint32 v = counts[i];
    s[tid] = v;
    __syncthreads();
    // Hillis-Steele inclusive scan in LDS
    for (int off = 1; off < SCAN_CHUNK; off <<= 1) {
        uint32 t = (tid >= off) ? s[tid - off] : 0u;
        __syncthreads();
        s[tid] += t;
        __syncthreads();
    }
    uint32 incl = s[tid];
    offsets[i] = incl - v;                      // exclusive within block
    if (tid == SCAN_CHUNK - 1) blockSums[blockIdx.x] = incl;
}

// ===========================================================================
// Phase 2b: exclusive scan of the 1024 block sums (single 1024-thread block)
// ===========================================================================
__global__ __launch_bounds__(SCAN_BLOCKS) void scan_sums_kernel(
    uint32* __restrict__ blockSums) {
    __shared__ uint32 s[SCAN_BLOCKS];
    const int tid = threadIdx.x;
    uint32 v = blockSums[tid];
    s[tid] = v;
    __syncthreads();
    for (int off = 1; off < SCAN_BLOCKS; off <<= 1) {
        uint32 t = (tid >= off) ? s[tid - off] : 0u;
        __syncthreads();
        s[tid] += t;
        __syncthreads();
    }
    blockSums[tid] = s[tid] - v;                // exclusive
}

// ===========================================================================
// Phase 2c: add block base to offsets; init fill cursors; emit fused
//           per-row descriptor {offset, count} for the pull kernel (one b64).
// ===========================================================================
__global__ __launch_bounds__(SCAN_CHUNK) void add_base_kernel(
    uint32* __restrict__ offsets,
    uint32* __restrict__ cursor,
    uint2*  __restrict__ rowdesc,
    const uint32* __restrict__ counts,
    const uint32* __restrict__ blockSums) {
    const int i = blockIdx.x * SCAN_CHUNK + threadIdx.x;
    uint32 o = offsets[i] + blockSums[blockIdx.x];
    offsets[i] = o;
    cursor[i]  = o;
    rowdesc[i] = make_uint2(o, counts[i]);
}

// ===========================================================================
// Phase 3: bucket fill  slotFrom[pos] = from_idx[e]  (1M returning u32 atomics)
// ===========================================================================
__global__ __launch_bounds__(256) void fill_kernel(const int* __restrict__ from_idx,
                                                   const int* __restrict__ to_idx,
                                                   uint32* __restrict__ cursor,
                                                   uint32* __restrict__ slotFrom,
                                                   int nnz) {
    int e = blockIdx.x * blockDim.x + threadIdx.x;
    if (e < nnz) {
        int t = to_idx[e];
        uint32 pos = atomicAdd(&cursor[t], 1u);
        slotFrom[pos] = (uint32)from_idx[e];
    }
}

// ===========================================================================
// Phase 4: pull. One wave32 per output row.
//   - one global_load_b64 for the row descriptor {offset, count}
//   - coalesced preload of the row's edge list (lane l -> edge l)
//   - one global_prefetch_b8 per gathered row head (in flight early)
//   - v_readlane broadcast (SGPR row id -> scalar base address, so the
//     gather is saddr-form global_load_b128 with scale_offset)
//   - 2-way unrolled gather loop: two b128 loads in flight per wait
//   - single global_store_b128 per lane writes the row (no RMW on d_out)
// Rows with zero edges store zeros, so no separate d_out zeroing pass needed.
// ===========================================================================
__global__ __launch_bounds__(256) void pull_kernel(
    const float*  __restrict__ x,
    const uint32* __restrict__ slotFrom,
    const uint2*  __restrict__ rowdesc,
    float* __restrict__ out) {

    const int lane = threadIdx.x & 31;
    const int row  = (blockIdx.x * blockDim.x + threadIdx.x) >> 5;  // < NUM_TO

    const uint2  dsc   = rowdesc[row];      // wave-uniform {offset, count}
    const uint32 start = dsc.x;
    const uint32 cnt   = dsc.y;

    float4 acc = make_float4(0.0f, 0.0f, 0.0f, 0.0f);

    for (uint32 base = 0; base < cnt; base += 32u) {
        uint32 rem = cnt - base;
        uint32 n   = rem < 32u ? rem : 32u;

        // one coalesced b32 load covers up to 32 edge ids for this row
        uint32 fv = 0u;
        if (lane < (int)n) {
            fv = slotFrom[start + base + (uint32)lane];
            // pull the head cacheline of each gathered row early
            __builtin_prefetch(x + (size_t)fv * D, 0, 3);
        }

        uint32 j = 0;
        for (; j + 2u <= n; j += 2u) {
            uint32 f0 = (uint32)__builtin_amdgcn_readlane((int)fv, (int)j);
            uint32 f1 = (uint32)__builtin_amdgcn_readlane((int)fv, (int)(j + 1u));
            const float4 v0 = ((const float4*)(x + (size_t)f0 * D))[lane];
            const float4 v1 = ((const float4*)(x + (size_t)f1 * D))[lane];
            acc.x += v0.x + v1.x;
            acc.y += v0.y + v1.y;
            acc.z += v0.z + v1.z;
            acc.w += v0.w + v1.w;
        }
        if (j < n) {
            uint32 f = (uint32)__builtin_amdgcn_readlane((int)fv, (int)j);
            const float4 v = ((const float4*)(x + (size_t)f * D))[lane];
            acc.x += v.x; acc.y += v.y; acc.z += v.z; acc.w += v.w;
        }
    }

    ((float4*)(out + (size_t)row * D))[lane] = acc;
}

// ===========================================================================
// Fallback (only if d_ws is too small): zero + direct f32-atomic scatter-add.
// ===========================================================================
__global__ __launch_bounds__(256) void zero_out_kernel(float4* __restrict__ out,
                                                       int n4) {
    int i = blockIdx.x * blockDim.x + threadIdx.x;
    if (i < n4) out[i] = make_float4(0.0f, 0.0f, 0.0f, 0.0f);
}

__global__ __launch_bounds__(256) void scatter_add_kernel(
    const float* __restrict__ x,
    const int*   __restrict__ from_idx,
    const int*   __restrict__ to_idx,
    float*       __restrict__ out,
    int nnz, int nwaves) {
    const int lane = threadIdx.x & 31;
    const int wave = (blockIdx.x * blockDim.x + threadIdx.x) >> 5;
    for (int e = wave; e < nnz; e += nwaves) {
        const int f = from_idx[e];
        const int t = to_idx[e];
        const float4 v = ((const float4*)(x + f * D))[lane];
        float* dst = out + t * D + lane * 4;
        atomicAdd(dst + 0, v.x);
        atomicAdd(dst + 1, v.y);
        atomicAdd(dst + 2, v.z);
        atomicAdd(dst + 3, v.w);
    }
}

// ===========================================================================
// Launch
// ===========================================================================
extern "C" void kernel_launch(void* const* d_in, const int* in_sizes, int n_in,
                              void* d_out, int out_size, void* d_ws, size_t ws_size,
                              hipStream_t stream) {
    const float* x        = (const float*)d_in[0];
    const int*   from_idx = (const int*)d_in[1];
    const int*   to_idx   = (const int*)d_in[2];
    float*       out      = (float*)d_out;
    const int    nnz      = in_sizes[1];

    // Workspace layout (u32 unless noted):
    //   counts | offsets | cursor | blockSums | slotFrom | rowdesc(uint2)
    const size_t u32_cnt = (size_t)(3 * NUM_TO + SCAN_BLOCKS + nnz);
    const size_t need    = u32_cnt * sizeof(uint32) + (size_t)NUM_TO * sizeof(uint2);

    if (ws_size >= need && nnz <= NNZ_EXP) {
        uint32* counts    = (uint32*)d_ws;
        uint32* offsets   = counts + NUM_TO;
        uint32* cursor    = offsets + NUM_TO;
        uint32* blockSums = cursor + NUM_TO;
        uint32* slotFrom  = blockSums + SCAN_BLOCKS;
        uint2*  rowdesc   = (uint2*)(slotFrom + nnz);   // 8B-aligned (u32_cnt even)

        // 0) zero counts
        zero_u32_kernel<<<(NUM_TO / 4 + 255) / 256, 256, 0, stream>>>(
            (uint4*)counts, NUM_TO / 4);
        // 1) histogram
        hist_kernel<<<(nnz + 255) / 256, 256, 0, stream>>>(to_idx, counts, nnz);
        // 2) two-level exclusive scan -> offsets, cursor, rowdesc
        block_scan_kernel<<<SCAN_BLOCKS, SCAN_CHUNK, 0, stream>>>(counts, offsets,
                                                                  blockSums);
        scan_sums_kernel<<<1, SCAN_BLOCKS, 0, stream>>>(blockSums);
        add_base_kernel<<<SCAN_BLOCKS, SCAN_CHUNK, 0, stream>>>(offsets, cursor,
                                                                rowdesc, counts,
                                                                blockSums);
        // 3) bucket fill
        fill_kernel<<<(nnz + 255) / 256, 256, 0, stream>>>(from_idx, to_idx,
                                                           cursor, slotFrom, nnz);
        // 4) pull: one wave per row, 8 waves/block -> NUM_TO/8 blocks
        pull_kernel<<<NUM_TO / 8, 256, 0, stream>>>(x, slotFrom, rowdesc, out);
    } else {
        // Fallback: direct atomic scatter-add
        const int n4 = out_size >> 2;
        zero_out_kernel<<<(n4 + 255) / 256, 256, 0, stream>>>((float4*)out, n4);
        const int blocks = 8192;
        const int nwaves = blocks * (256 / 32);
        scatter_add_kernel<<<blocks, 256, 0, stream>>>(x, from_idx, to_idx, out,
                                                       nnz, nwaves);
    }
}